// SNN_Model_73409581023681
// MI455X (gfx1250) — compile-verified
//
#include <hip/hip_runtime.h>
#include <math.h>

// ---- problem constants (from reference setup_inputs) ----
#define B_SZ   4096
#define INDIM  784
#define H1     512
#define H2     10
#define H2P    16        // layer-2 width padded to one WMMA tile
#define WINS   10
#define TAU_W  40.0f
#define THRESH 0.35f

typedef float v2f __attribute__((ext_vector_type(2)));
typedef float v8f __attribute__((ext_vector_type(8)));

// fp32 WMMA: D(16x16) = A(16x4) * B(4x16) + C   (exact fp32, matches reference numerics)
__device__ __forceinline__ v8f wmma4(v2f a, v2f b, v8f c) {
  return __builtin_amdgcn_wmma_f32_16x16x4_f32(
      /*neg_a=*/false, a, /*neg_b=*/false, b,
      /*c_mod=*/(short)0, c, /*reuse_a=*/false, /*reuse_b=*/false);
}

// ---------------- elementwise helpers ----------------
__global__ void snn_zero(float* p, int n) {
  int i = blockIdx.x * blockDim.x + threadIdx.x;
  if (i < n) p[i] = 0.0f;
}
__global__ void snn_copy(float* d, const float* s, int n) {
  int i = blockIdx.x * blockDim.x + threadIdx.x;
  if (i < n) d[i] = s[i];
}
// hb2 padded (512x16) from input (512x10)
__global__ void snn_init_hb2p(float* hb2p, const float* hebb2) {
  int i = blockIdx.x * blockDim.x + threadIdx.x;
  if (i < H1 * H2P) {
    int k = i >> 4, n = i & 15;
    hb2p[i] = (n < H2) ? hebb2[k * H2 + n] : 0.0f;
  }
}
// extract hb2 (512x10) from padded
__global__ void snn_hb2_out(float* out, const float* hb2p) {
  int i = blockIdx.x * blockDim.x + threadIdx.x;
  if (i < H1 * H2) {
    int k = i / H2, n = i - k * H2;
    out[i] = hb2p[k * H2P + n];
  }
}
// Weff1[k,n] = W1[n,k] + alpha1 * hb1[k,n]   (784x512)
__global__ void snn_weff1(float* W, const float* W1, const float* hb1,
                          const float* alpha1) {
  int i = blockIdx.x * blockDim.x + threadIdx.x;
  if (i < INDIM * H1) {
    int k = i >> 9, n = i & (H1 - 1);
    W[i] = W1[n * INDIM + k] + alpha1[0] * hb1[i];
  }
}
// LIF layer 1 + spike sum + post = tanh(h/THRESH - eta1)
__global__ void snn_lif1(const float* __restrict__ h1mem, float* v1, float* s1,
                         float* post1, float* h1sum, const float* __restrict__ eta1) {
  int i = blockIdx.x * blockDim.x + threadIdx.x;
  if (i < B_SZ * H1) {
    int n = i & (H1 - 1);
    float h = h1mem[i];
    float v = 0.5f * (v1[i] + h);                 // v += (h - v)/tau, tau = 2
    float s = (v >= 1.0f) ? 1.0f : 0.0f;
    v1[i] = (1.0f - s) * v;                       // hard reset
    s1[i] = s;
    h1sum[i] += s;
    post1[i] = tanhf(h * (1.0f / THRESH) - eta1[n]);
  }
}
// LIF layer 2 (padded to 16) + h2_sum accumulate + h2_spk (last write wins)
__global__ void snn_lif2(const float* __restrict__ h2mem, float* v2, float* post2,
                         float* h2sum, float* h2spk, const float* __restrict__ eta2) {
  int i = blockIdx.x * blockDim.x + threadIdx.x;
  if (i < B_SZ * H2P) {
    int n = i & 15, b = i >> 4;
    float h = h2mem[i];
    float v = 0.5f * (v2[i] + h);
    float s = (v >= 1.0f) ? 1.0f : 0.0f;
    v2[i] = (1.0f - s) * v;
    float e = (n < H2) ? eta2[n] : 0.0f;
    post2[i] = tanhf(h * (1.0f / THRESH) - e);
    if (n < H2) {
      h2sum[b * H2 + n] += s;
      h2spk[b * H2 + n] = s;                      // overwritten each step; last = step 9
    }
  }
}

// ---------------- WMMA GEMM kernels ----------------
// C[4096,512] = (scale*X[4096,784]) @ W[784,512] + b[512]
// 8 waves/block; each wave owns a 32x64 tile: B fragments reused 2x, A fragments 4x.
__global__ void snn_gemm1(const float* __restrict__ X, const float* __restrict__ W,
                          const float* __restrict__ bias, float* __restrict__ C,
                          float scale) {
  const int K = INDIM, N = H1;
  int wave = (blockIdx.x * blockDim.x + threadIdx.x) >> 5;
  int lane = threadIdx.x & 31;
  int half = lane >> 4, lr = lane & 15;
  int m0 = (wave >> 3) << 5;          // 128 row groups of 32
  int n0 = (wave & 7) << 6;           // 8 col groups of 64
  v8f acc[2][4] = {};
  const float* xr0 = X + (size_t)(m0 + lr) * K + 2 * half;
  const float* xr1 = xr0 + (size_t)16 * K;
  for (int k = 0; k < K; k += 4) {
    v2f a0, a1;
    a0.x = xr0[k] * scale; a0.y = xr0[k + 1] * scale;
    a1.x = xr1[k] * scale; a1.y = xr1[k + 1] * scale;
    const float* w0 = W + (size_t)(k + 2 * half) * N + n0 + lr;
    const float* w1 = w0 + N;
    v2f b[4];
#pragma unroll
    for (int j = 0; j < 4; ++j) { b[j].x = w0[16 * j]; b[j].y = w1[16 * j]; }
#pragma unroll
    for (int j = 0; j < 4; ++j) {
      acc[0][j] = wmma4(a0, b[j], acc[0][j]);
      acc[1][j] = wmma4(a1, b[j], acc[1][j]);
    }
  }
#pragma unroll
  for (int mi = 0; mi < 2; ++mi)
#pragma unroll
    for (int j = 0; j < 4; ++j)
#pragma unroll
      for (int r = 0; r < 8; ++r) {
        int m = m0 + 16 * mi + r + 8 * half;
        int n = n0 + 16 * j + lr;
        C[(size_t)m * N + n] = acc[mi][j][r] + bias[n];
      }
}

// hb1[784,512] = clip(0.95*hb1 + (1/B) * (scale*beta1[i]*x[b,i])^T @ post1[b,j], -4, 4)
// Each wave owns a 16x32 tile: A fragment reused 2x.
__global__ void snn_hebb1(const float* __restrict__ X, const float* __restrict__ beta1,
                          const float* __restrict__ post, float* __restrict__ hb,
                          float scale) {
  int wave = (blockIdx.x * blockDim.x + threadIdx.x) >> 5;
  int lane = threadIdx.x & 31;
  int half = lane >> 4, lr = lane & 15;
  int i0 = (wave >> 4) << 4;          // 49 row tiles (784/16)
  int j0 = (wave & 15) << 5;          // 16 col groups of 32
  float bsc = beta1[i0 + lr] * scale;
  v8f acc0 = {}, acc1 = {};
  const float* xp = X + (size_t)(2 * half) * INDIM + i0 + lr;
  const float* pp = post + (size_t)(2 * half) * H1 + j0 + lr;
  for (int k = 0; k < B_SZ; k += 4) {
    v2f a, b0, b1;
    a.x = xp[(size_t)k * INDIM] * bsc;
    a.y = xp[(size_t)(k + 1) * INDIM] * bsc;
    b0.x = pp[(size_t)k * H1];      b0.y = pp[(size_t)(k + 1) * H1];
    b1.x = pp[(size_t)k * H1 + 16]; b1.y = pp[(size_t)(k + 1) * H1 + 16];
    acc0 = wmma4(a, b0, acc0);
    acc1 = wmma4(a, b1, acc1);
  }
  const float invB = 1.0f / (float)B_SZ;
#pragma unroll
  for (int r = 0; r < 8; ++r) {
    int i = i0 + r + 8 * half;
    float v0 = 0.95f * hb[(size_t)i * H1 + j0 + lr] + acc0[r] * invB;
    float v1 = 0.95f * hb[(size_t)i * H1 + j0 + 16 + lr] + acc1[r] * invB;
    hb[(size_t)i * H1 + j0 + lr]      = fminf(fmaxf(v0, -4.0f), 4.0f);
    hb[(size_t)i * H1 + j0 + 16 + lr] = fminf(fmaxf(v1, -4.0f), 4.0f);
  }
}

// C[4096,16] = S1[4096,512] @ (W2^T + alpha2*hb2p) + b2(padded)
// Effective weight (512x16 = 32KB) is built pair-interleaved in LDS by the block
// (fuses the old weff2 kernel); all 8 waves share it via ds_load_b64.
__global__ void snn_gemm2(const float* __restrict__ S1, const float* __restrict__ W2,
                          const float* __restrict__ hb2p, const float* __restrict__ alpha2,
                          const float* __restrict__ b2, float* __restrict__ C) {
  const int K = H1, N = H2P;
  __shared__ v2f lds2[(K / 2) * N];   // lds2[(k/2)*16 + n] = {Weff[k,n], Weff[k+1,n]}
  float a2 = alpha2[0];
  for (int p = threadIdx.x; p < (K / 2) * N; p += blockDim.x) {
    int k0 = (p >> 4) << 1, n = p & 15;
    v2f w;
    if (n < H2) {
      w.x = W2[n * K + k0]     + a2 * hb2p[k0 * N + n];
      w.y = W2[n * K + k0 + 1] + a2 * hb2p[(k0 + 1) * N + n];
    } else {
      w.x = 0.0f; w.y = 0.0f;
    }
    lds2[p] = w;
  }
  __syncthreads();

  int wave = (blockIdx.x * blockDim.x + threadIdx.x) >> 5;
  int lane = threadIdx.x & 31;
  int half = lane >> 4, lr = lane & 15;
  int m0 = wave << 4;                 // 256 row tiles
  v8f acc = {};
  const float* srow = S1 + (size_t)(m0 + lr) * K + 2 * half;
  for (int k = 0; k < K; k += 4) {
    v2f a, b;
    a.x = srow[k];
    a.y = srow[k + 1];
    b = lds2[((k + 2 * half) >> 1) * 16 + lr];
    acc = wmma4(a, b, acc);
  }
  float bb = (lr < H2) ? b2[lr] : 0.0f;
#pragma unroll
  for (int r = 0; r < 8; ++r) {
    int m = m0 + r + 8 * half;
    C[(size_t)m * N + lr] = acc[r] + bb;
  }
}

// hb2p[512,16] = clip(0.95*hb2p + (1/B) * (beta2[i]*s1[b,i])^T @ post2[b,j], -4, 4)
__global__ void snn_hebb2(const float* __restrict__ S1, const float* __restrict__ beta2,
                          const float* __restrict__ post2, float* __restrict__ hb) {
  int wave = (blockIdx.x * blockDim.x + threadIdx.x) >> 5;
  int lane = threadIdx.x & 31;
  int half = lane >> 4, lr = lane & 15;
  int i0 = wave << 4;                 // 32 row tiles (512/16)
  float bsc = beta2[i0 + lr];
  v8f acc = {};
  for (int k = 0; k < B_SZ; k += 4) {
    v2f a, b;
    a.x = S1[(size_t)(k + 2 * half) * H1 + i0 + lr] * bsc;
    a.y = S1[(size_t)(k + 2 * half + 1) * H1 + i0 + lr] * bsc;
    b.x = post2[(size_t)(k + 2 * half) * H2P + lr];
    b.y = post2[(size_t)(k + 2 * half + 1) * H2P + lr];
    acc = wmma4(a, b, acc);
  }
  const float invB = 1.0f / (float)B_SZ;
#pragma unroll
  for (int r = 0; r < 8; ++r) {
    int i = i0 + r + 8 * half;
    float v = 0.95f * hb[(size_t)i * H2P + lr] + acc[r] * invB;
    hb[(size_t)i * H2P + lr] = fminf(fmaxf(v, -4.0f), 4.0f);
  }
}

// ---------------- host orchestration ----------------
extern "C" void kernel_launch(void* const* d_in, const int* in_sizes, int n_in,
                              void* d_out, int out_size, void* d_ws, size_t ws_size,
                              hipStream_t stream) {
  const float* x        = (const float*)d_in[0];
  const float* hebb1_in = (const float*)d_in[1];
  const float* hebb2_in = (const float*)d_in[2];
  const float* W1       = (const float*)d_in[3];
  const float* b1       = (const float*)d_in[4];
  const float* W2       = (const float*)d_in[5];
  const float* b2       = (const float*)d_in[6];
  const float* alpha1   = (const float*)d_in[7];
  const float* alpha2   = (const float*)d_in[8];
  const float* eta1     = (const float*)d_in[9];
  const float* eta2     = (const float*)d_in[10];
  const float* beta1    = (const float*)d_in[11];
  const float* beta2    = (const float*)d_in[12];
  float* out = (float*)d_out;
  float* ws  = (float*)d_ws;

  // workspace layout (floats)
  size_t o = 0;
  float* hb1   = ws + o; o += (size_t)INDIM * H1;
  float* hb2p  = ws + o; o += (size_t)H1 * H2P;
  float* weff1 = ws + o; o += (size_t)INDIM * H1;
  float* h1mem = ws + o; o += (size_t)B_SZ * H1;
  float* post1 = ws + o; o += (size_t)B_SZ * H1;
  float* s1    = ws + o; o += (size_t)B_SZ * H1;
  float* v1    = ws + o; o += (size_t)B_SZ * H1;
  float* h2mem = ws + o; o += (size_t)B_SZ * H2P;
  float* post2 = ws + o; o += (size_t)B_SZ * H2P;
  float* v2    = ws + o; o += (size_t)B_SZ * H2P;

  // output layout (reference return order, flat)
  float* out_h2spk = out;
  float* out_h1sum = out_h2spk + (size_t)B_SZ * H2;
  float* out_h2sum = out_h1sum + (size_t)B_SZ * H1;
  float* out_hebb1 = out_h2sum + (size_t)B_SZ * H2;
  float* out_hebb2 = out_hebb1 + (size_t)INDIM * H1;
  float* out_eta1  = out_hebb2 + (size_t)H1 * H2;
  float* out_eta2  = out_eta1 + H1;

  auto g = [](int n) { return (n + 255) / 256; };

  // re-initialize all state from inputs every call (deterministic)
  snn_zero<<<g(B_SZ * H1), 256, 0, stream>>>(v1, B_SZ * H1);
  snn_zero<<<g(B_SZ * H2P), 256, 0, stream>>>(v2, B_SZ * H2P);
  snn_zero<<<g(B_SZ * H1), 256, 0, stream>>>(out_h1sum, B_SZ * H1);
  snn_zero<<<g(B_SZ * H2), 256, 0, stream>>>(out_h2sum, B_SZ * H2);
  snn_copy<<<g(INDIM * H1), 256, 0, stream>>>(hb1, hebb1_in, INDIM * H1);
  snn_init_hb2p<<<g(H1 * H2P), 256, 0, stream>>>(hb2p, hebb2_in);
  snn_copy<<<g(H1), 256, 0, stream>>>(out_eta1, eta1, H1);
  snn_copy<<<g(H2), 256, 0, stream>>>(out_eta2, eta2, H2);

  for (int t = 0; t < WINS; ++t) {
    float scale = expf(-(float)t / TAU_W);
    // layer 1
    snn_weff1<<<g(INDIM * H1), 256, 0, stream>>>(weff1, W1, hb1, alpha1);
    snn_gemm1<<<128, 256, 0, stream>>>(x, weff1, b1, h1mem, scale);          // 1024 waves, 32x64/wave
    snn_lif1<<<g(B_SZ * H1), 256, 0, stream>>>(h1mem, v1, s1, post1, out_h1sum, eta1);
    snn_hebb1<<<98, 256, 0, stream>>>(x, beta1, post1, hb1, scale);          // 784 waves, 16x32/wave
    // layer 2 (weff2 fused into gemm2 via LDS)
    snn_gemm2<<<32, 256, 0, stream>>>(s1, W2, hb2p, alpha2, b2, h2mem);      // 256 waves
    snn_lif2<<<g(B_SZ * H2P), 256, 0, stream>>>(h2mem, v2, post2, out_h2sum, out_h2spk, eta2);
    snn_hebb2<<<4, 256, 0, stream>>>(s1, beta2, post2, hb2p);                // 32 waves
  }

  snn_copy<<<g(INDIM * H1), 256, 0, stream>>>(out_hebb1, hb1, INDIM * H1);
  snn_hb2_out<<<g(H1 * H2), 256, 0, stream>>>(out_hebb2, hb2p);
}